// SMFNet_64347200028731
// MI455X (gfx1250) — compile-verified
//
#include <hip/hip_runtime.h>
#include <hip/hip_bf16.h>

typedef float v2f __attribute__((ext_vector_type(2)));
typedef float v8f __attribute__((ext_vector_type(8)));

#define NROW 4096   // N
#define KDIM 512    // D_IN == H1
#define BM 64
#define BN 64
#define BK 32
#define LSTRIDE 34  // padded LDS row stride (words): 34*i mod 64 distinct for i=0..15

// C = relu(A[M,K] @ B[K,N] + bias[N]) using V_WMMA_F32_16X16X4_F32.
// Block: 256 threads = 8 wave32s, arranged 4(M) x 2(N); wave tile 16x32 (2 accumulators).
// Tiles staged via double-buffered GLOBAL_LOAD_ASYNC_TO_LDS (ASYNCcnt), B transposed in LDS.
__global__ __launch_bounds__(256)
void gemm_bias_relu_wmma(const float* __restrict__ A, const float* __restrict__ B,
                         const float* __restrict__ bias, float* __restrict__ C,
                         int M, int N, int K)
{
    __shared__ float As[2][BM * LSTRIDE];   // [m][k], 8704 B per buffer
    __shared__ float Bt[2][BN * LSTRIDE];   // [n][k] (transposed), 8704 B per buffer

    const int t    = threadIdx.x;
    const int wave = t >> 5;
    const int lane = t & 31;
    const int wm   = wave >> 1;          // 0..3
    const int wn   = wave & 1;           // 0..1
    const int bm   = blockIdx.y * BM;
    const int bn   = blockIdx.x * BN;

    const int lm = lane & 15;            // M/N index within 16
    const int kh = (lane >> 4) * 2;      // K sub-offset: 0 or 2

    // B fill pattern: lanes span consecutive n at fixed k (coalesced global reads,
    // element-granular async copy performs the transpose into LDS).
    const int bn_loc  = t & 63;          // 0..63
    const int bk_base = t >> 6;          // 0..3

    v8f acc0 = {};
    v8f acc1 = {};

    // Issue async global->LDS copies for the k-tile at kb into buffer `buf`.
    auto load_tile_async = [&](int kb, int buf) {
        // ---- A tile: BM x BK floats as 4 async B64 copies per thread
        #pragma unroll
        for (int p = 0; p < 4; ++p) {
            int f  = t + p * 256;            // b64 chunk 0..1023
            int m  = f >> 4;                 // 16 chunks per row
            int k2 = (f & 15) << 1;
            const float* gp = &A[(size_t)(bm + m) * K + kb + k2];
            unsigned loff = (unsigned)(unsigned long long)(&As[buf][m * LSTRIDE + k2]);
            asm volatile("global_load_async_to_lds_b64 %0, %1, off"
                         :: "v"(loff), "v"((unsigned long long)gp) : "memory");
        }
        // ---- B tile: BK x BN floats as 8 async B32 copies per thread (transposed)
        #pragma unroll
        for (int p = 0; p < 8; ++p) {
            int k = bk_base + (p << 2);      // 0..31
            const float* gp = &B[(size_t)(kb + k) * N + bn + bn_loc];
            unsigned loff = (unsigned)(unsigned long long)(&Bt[buf][bn_loc * LSTRIDE + k]);
            asm volatile("global_load_async_to_lds_b32 %0, %1, off"
                         :: "v"(loff), "v"((unsigned long long)gp) : "memory");
        }
    };

    load_tile_async(0, 0);
    asm volatile("s_wait_asynccnt 0x0" ::: "memory");
    __syncthreads();

    int buf = 0;
    for (int kb = 0; kb < K; kb += BK) {
        if (kb + BK < K)
            load_tile_async(kb + BK, buf ^ 1);   // overlap with compute below

        #pragma unroll
        for (int kk = 0; kk < BK; kk += 4) {
            const int k0 = kk + kh;
            // A frag: element (M = wm*16 + lm, K = k0 + v)  -> contiguous b64 in LDS
            v2f a = *(const v2f*)&As[buf][(wm * 16 + lm) * LSTRIDE + k0];
            // B frags: element (N, K = k0 + v) -> contiguous b64 in transposed LDS
            v2f b0 = *(const v2f*)&Bt[buf][(wn * 32 + lm) * LSTRIDE + k0];
            v2f b1 = *(const v2f*)&Bt[buf][(wn * 32 + 16 + lm) * LSTRIDE + k0];
            acc0 = __builtin_amdgcn_wmma_f32_16x16x4_f32(
                false, a, false, b0, (short)0, acc0, false, false);
            acc1 = __builtin_amdgcn_wmma_f32_16x16x4_f32(
                false, a, false, b1, (short)0, acc1, false, false);
        }

        // Own async copies into buf^1 must land before the barrier; barrier then
        // guarantees everyone finished reading `buf` before it is refilled.
        asm volatile("s_wait_asynccnt 0x0" ::: "memory");
        __syncthreads();
        buf ^= 1;
    }

    // epilogue: C/D layout -> M = v + (lane>>4)*8, N = lane%16
    const int row0 = bm + wm * 16 + (lane >> 4) * 8;
    const int c0   = bn + wn * 32 + lm;
    const float bias0 = bias[c0];
    const float bias1 = bias[c0 + 16];
    #pragma unroll
    for (int v = 0; v < 8; ++v) {
        float x0 = acc0[v] + bias0;
        float x1 = acc1[v] + bias1;
        C[(size_t)(row0 + v) * N + c0]      = x0 > 0.f ? x0 : 0.f;
        C[(size_t)(row0 + v) * N + c0 + 16] = x1 > 0.f ? x1 : 0.f;
    }
}

// For each row i: d[i] = relu(H[i] . W2[:,i] + b2[i]),
//                 s[i] = relu(H[i] . W2[:,(i+1)%N] + b2[(i+1)%N])
// One wave32 per row; shfl_xor tree reduction. W2 (24 MB) stays L2-resident (192 MB L2).
__global__ __launch_bounds__(256)
void fdiag_kernel(const float* __restrict__ H, const float* __restrict__ W2,
                  const float* __restrict__ b2, float* __restrict__ dvec,
                  float* __restrict__ svec)
{
    const int wave = threadIdx.x >> 5;
    const int lane = threadIdx.x & 31;
    const int i = blockIdx.x * 8 + wave;       // row 0..4095
    const int j = (i + 1) & (NROW - 1);

    float ad = 0.f, as = 0.f;
    #pragma unroll 4
    for (int k = lane; k < KDIM; k += 32) {
        float h = H[(size_t)i * KDIM + k];
        ad += h * W2[(size_t)k * NROW + i];
        as += h * W2[(size_t)k * NROW + j];
    }
    #pragma unroll
    for (int off = 16; off > 0; off >>= 1) {
        ad += __shfl_xor(ad, off, 32);
        as += __shfl_xor(as, off, 32);
    }
    if (lane == 0) {
        float d = ad + b2[i];
        float s = as + b2[j];
        dvec[i] = d > 0.f ? d : 0.f;
        svec[i] = s > 0.f ? s : 0.f;
    }
}

// V'[i] = d[i]*V[i] + s[i]*V[(i+1)%N]  (chord-masked aggregation), fully coalesced.
__global__ __launch_bounds__(256)
void chord_agg_kernel(const float* __restrict__ V, const float* __restrict__ dvec,
                      const float* __restrict__ svec, float* __restrict__ Vout)
{
    const int idx = blockIdx.x * 256 + threadIdx.x;  // 0 .. N*KDIM-1
    const int i = idx >> 9;                          // / 512
    const int c = idx & (KDIM - 1);
    const int j = (i + 1) & (NROW - 1);
    Vout[idx] = dvec[i] * V[idx] + svec[i] * V[(size_t)j * KDIM + c];
}

extern "C" void kernel_launch(void* const* d_in, const int* in_sizes, int n_in,
                              void* d_out, int out_size, void* d_ws, size_t ws_size,
                              hipStream_t stream) {
    const float* X   = (const float*)d_in[0];   // [4096,512]
    const float* gW  = (const float*)d_in[1];   // [2,512,512]
    const float* gb  = (const float*)d_in[2];   // [2,512]
    const float* fW1 = (const float*)d_in[3];   // [3,1,512,512]
    const float* fb1 = (const float*)d_in[4];   // [3,1,512]
    const float* fW2 = (const float*)d_in[5];   // [3,512,4096]
    const float* fb2 = (const float*)d_in[6];   // [3,4096]
    float* out = (float*)d_out;                 // [4096,512]

    float* ws = (float*)d_ws;
    const size_t VSZ = (size_t)NROW * KDIM;     // 2M floats
    float* Va   = ws;
    float* Vb   = ws + VSZ;
    float* Hh   = ws + 2 * VSZ;
    float* dvec = ws + 3 * VSZ;
    float* svec = dvec + NROW;

    dim3 gemm_grid(KDIM / BN, NROW / BM);       // (8, 64)
    dim3 blk(256);

    // g MLP: V = relu(relu(X@gW0+gb0)@gW1+gb1)
    gemm_bias_relu_wmma<<<gemm_grid, blk, 0, stream>>>(X,  gW,              gb,        Va, NROW, KDIM, KDIM);
    gemm_bias_relu_wmma<<<gemm_grid, blk, 0, stream>>>(Va, gW + KDIM*KDIM, gb + KDIM, Vb, NROW, KDIM, KDIM);

    const float* Vcur = Vb;
    float* Vnext[3] = { Va, Vb, out };
    for (int m = 0; m < 3; ++m) {
        gemm_bias_relu_wmma<<<gemm_grid, blk, 0, stream>>>(
            X, fW1 + (size_t)m * KDIM * KDIM, fb1 + (size_t)m * KDIM, Hh, NROW, KDIM, KDIM);
        fdiag_kernel<<<NROW / 8, blk, 0, stream>>>(
            Hh, fW2 + (size_t)m * KDIM * NROW, fb2 + (size_t)m * NROW, dvec, svec);
        chord_agg_kernel<<<(NROW * KDIM) / 256, blk, 0, stream>>>(Vcur, dvec, svec, Vnext[m]);
        Vcur = Vnext[m];
    }
}